// GCN_33998961116038
// MI455X (gfx1250) — compile-verified
//
#include <hip/hip_runtime.h>

typedef float v2f __attribute__((ext_vector_type(2)));
typedef float v8f __attribute__((ext_vector_type(8)));

#define N_NODES 100000
#define N_EDGES 1600000

// ---------------------------------------------------------------------------
// Elementwise helpers
// ---------------------------------------------------------------------------
__global__ void gcn_fill_zero(float* __restrict__ p, int n) {
    int i = blockIdx.x * blockDim.x + threadIdx.x;
    if (i < n) p[i] = 0.0f;
}

__global__ void gcn_degree(const int* __restrict__ col, float* __restrict__ deg, int e) {
    int i = blockIdx.x * blockDim.x + threadIdx.x;
    if (i < e) atomicAdd(&deg[col[i]], 1.0f);
}

__global__ void gcn_dinv(float* __restrict__ deg, int n) {
    int i = blockIdx.x * blockDim.x + threadIdx.x;
    if (i < n) {
        float d = deg[i];
        deg[i] = (d > 0.0f) ? (1.0f / sqrtf(d)) : 0.0f;   // deg^-0.5, else 0
    }
}

__global__ void gcn_norm(const int* __restrict__ row, const int* __restrict__ col,
                         const float* __restrict__ dinv, float* __restrict__ norm, int e) {
    int i = blockIdx.x * blockDim.x + threadIdx.x;
    if (i < e) norm[i] = dinv[row[i]] * dinv[col[i]];
}

// out[n][f] = b[f]  (accumulator seeded with bias; scatter-add lands on top)
template <int F>
__global__ void gcn_bias_init(float* __restrict__ out, const float* __restrict__ b, int total) {
    int i = blockIdx.x * blockDim.x + threadIdx.x;
    if (i < total) out[i] = b[i & (F - 1)];
}

// ---------------------------------------------------------------------------
// Dense GEMM via V_WMMA_F32_16X16X4_F32.  One wave computes one 16x16 tile of
// out = (RELU ? max(A,0) : A) @ W,  A:[nRows,128] row-major, W:[128,F] row-major.
// ---------------------------------------------------------------------------
template <int F, bool RELU>
__global__ __launch_bounds__(256) void gcn_gemm_wmma(const float* __restrict__ A,
                                                     const float* __restrict__ W,
                                                     float* __restrict__ out, int nRows) {
    constexpr int TILES_N = F / 16;
    const int wave = (int)((blockIdx.x * blockDim.x + threadIdx.x) >> 5);
    const int lane = threadIdx.x & 31;
    const int tm = wave / TILES_N;
    const int tn = wave - tm * TILES_N;
    if (tm * 16 >= nRows) return;

    const int mn    = lane & 15;   // row of A frag / col of B,D frags
    const int khalf = lane >> 4;   // which K pair this half-wave owns

    const float* arow = A + (size_t)(tm * 16 + mn) * 128;
    const float* wcol = W + (size_t)(tn * 16 + mn);

    v8f acc = {};
#pragma unroll
    for (int k = 0; k < 128; k += 4) {
        const int kb = k + 2 * khalf;
        // A 16x4 frag: lane holds (row mn, K = kb, kb+1) -> contiguous float2
        float a0 = arow[kb];
        float a1 = arow[kb + 1];
        if (RELU) { a0 = fmaxf(a0, 0.0f); a1 = fmaxf(a1, 0.0f); }
        v2f a; a.x = a0; a.y = a1;
        // B 4x16 frag: VGPR0 = K-rows {0,2}, VGPR1 = K-rows {1,3}
        v2f b;
        b.x = wcol[(size_t)kb * F];
        b.y = wcol[(size_t)(kb + 1) * F];
        acc = __builtin_amdgcn_wmma_f32_16x16x4_f32(
            /*neg_a=*/false, a, /*neg_b=*/false, b,
            /*c_mod=*/(short)0, acc, /*reuse_a=*/false, /*reuse_b=*/false);
    }

    // D 16x16: VGPR v holds rows {v, v+8} split across lane halves
    float* orow = out + (size_t)(tm * 16 + khalf * 8) * F + (size_t)(tn * 16 + mn);
#pragma unroll
    for (int v = 0; v < 8; ++v)
        orow[(size_t)v * F] = acc[v];
}

// ---------------------------------------------------------------------------
// Edge-parallel scatter-add:  out[col[e]][f] += t[row[e]][f] * norm[e]
// Block of 256 threads covers 256/F edges; lanes within an edge are coalesced.
// ---------------------------------------------------------------------------
template <int F>
__global__ __launch_bounds__(256) void gcn_scatter_add(const float* __restrict__ t,
                                                       const int* __restrict__ row,
                                                       const int* __restrict__ col,
                                                       const float* __restrict__ norm,
                                                       float* __restrict__ out) {
    constexpr int EPB = 256 / F;
    const int e = blockIdx.x * EPB + (threadIdx.x >> (F == 128 ? 7 : 6));
    const int f = threadIdx.x & (F - 1);
    const int r = row[e];
    const int c = col[e];
    const float v = t[(size_t)r * F + f] * norm[e];
    atomicAdd(&out[(size_t)c * F + f], v);
}

// ---------------------------------------------------------------------------
extern "C" void kernel_launch(void* const* d_in, const int* in_sizes, int n_in,
                              void* d_out, int out_size, void* d_ws, size_t ws_size,
                              hipStream_t stream) {
    const float* x   = (const float*)d_in[0];
    const int*   ei  = (const int*)d_in[1];     // [2, E] flattened, int32
    const float* W1  = (const float*)d_in[2];
    const float* b1  = (const float*)d_in[3];
    const float* W2  = (const float*)d_in[4];
    const float* b2  = (const float*)d_in[5];
    const float* W3  = (const float*)d_in[6];
    const float* b3  = (const float*)d_in[7];
    float*       out = (float*)d_out;

    const int* row = ei;
    const int* col = ei + N_EDGES;

    // Workspace partition (~109 MB):
    float* dinv = (float*)d_ws;                       // N
    float* norm = dinv + N_NODES;                     // E
    float* t    = norm + N_EDGES;                     // N*128 (GEMM output)
    float* h    = t + (size_t)N_NODES * 128;          // N*128 (aggregated activations)

    const int TB = 256;

    // norm[e] = deg^-0.5[row] * deg^-0.5[col]
    gcn_fill_zero<<<(N_NODES + TB - 1) / TB, TB, 0, stream>>>(dinv, N_NODES);
    gcn_degree<<<(N_EDGES + TB - 1) / TB, TB, 0, stream>>>(col, dinv, N_EDGES);
    gcn_dinv<<<(N_NODES + TB - 1) / TB, TB, 0, stream>>>(dinv, N_NODES);
    gcn_norm<<<(N_EDGES + TB - 1) / TB, TB, 0, stream>>>(row, col, dinv, norm, N_EDGES);

    const int gemm128Blocks = (N_NODES / 16) * (128 / 16) / 8;   // 8 waves per block
    const int gemm64Blocks  = (N_NODES / 16) * (64 / 16) / 8;
    const int scat128Blocks = N_EDGES / (256 / 128);
    const int scat64Blocks  = N_EDGES / (256 / 64);

    // Layer 1: t = x @ W1 ; h = b1 + scatter(t) ; relu folded into next GEMM
    gcn_gemm_wmma<128, false><<<gemm128Blocks, TB, 0, stream>>>(x, W1, t, N_NODES);
    gcn_bias_init<128><<<(N_NODES * 128) / TB, TB, 0, stream>>>(h, b1, N_NODES * 128);
    gcn_scatter_add<128><<<scat128Blocks, TB, 0, stream>>>(t, row, col, norm, h);

    // Layer 2: t = relu(h) @ W2 ; h = b2 + scatter(t)
    gcn_gemm_wmma<128, true><<<gemm128Blocks, TB, 0, stream>>>(h, W2, t, N_NODES);
    gcn_bias_init<128><<<(N_NODES * 128) / TB, TB, 0, stream>>>(h, b2, N_NODES * 128);
    gcn_scatter_add<128><<<scat128Blocks, TB, 0, stream>>>(t, row, col, norm, h);

    // Layer 3: t = relu(h) @ W3 ; out = b3 + scatter(t)
    gcn_gemm_wmma<64, true><<<gemm64Blocks, TB, 0, stream>>>(h, W3, t, N_NODES);
    gcn_bias_init<64><<<(N_NODES * 64) / TB, TB, 0, stream>>>(out, b3, N_NODES * 64);
    gcn_scatter_add<64><<<scat64Blocks, TB, 0, stream>>>(t, row, col, norm, out);
}